// UniformAug_8418135900825
// MI455X (gfx1250) — compile-verified
//
#include <hip/hip_runtime.h>
#include <math.h>

typedef float v2f __attribute__((ext_vector_type(2)));
typedef float v4f __attribute__((ext_vector_type(4)));
typedef float v8f __attribute__((ext_vector_type(8)));

#define IMG_H 224
#define IMG_W 224
#define IMG_C 3
#define HW    (IMG_H * IMG_W)
#define PXT   4                      // pixels per thread (b128 store)
#define XQ    (IMG_W / PXT)          // 56
#define PIX_PER_B (IMG_H * XQ)       // 12544
#define BLK   256
#define NBLK  (PIX_PER_B / BLK)      // 49 (exact)

// ---------------------------------------------------------------------------
// Wave-level 3x3 matmul using V_WMMA_F32_16X16X4_F32 (3x3 embedded in 16x4*4x16).
// t,g are wave-uniform 9-element row-major 3x3 matrices; r = t @ g.
// A layout (16x4 f32): lanes 0-15 -> M=lane, VGPR{0,1}=K{0,1}; lanes 16-31 -> K{2,3}.
// B layout (4x16 f32): mirrored (N on lanes, K across VGPRs/halves).
// C/D layout (16x16 f32): element (m,n), m<8 -> VGPR m, lane n.
// ---------------------------------------------------------------------------
__device__ __forceinline__ float mat_el(const float* M, int r, int c) {
  return (r < 3 && c < 3) ? M[r * 3 + c] : 0.0f;
}

__device__ __forceinline__ void wave_mm3(const float* t, const float* g, float* r) {
  const int lane = threadIdx.x & 31;
  const int ml = lane & 15;
  const bool hi = lane >= 16;
  v2f a, b;
  a.x = mat_el(t, ml, hi ? 2 : 0);
  a.y = mat_el(t, ml, hi ? 3 : 1);
  b.x = mat_el(g, hi ? 2 : 0, ml);
  b.y = mat_el(g, hi ? 3 : 1, ml);
  v8f c = {};
  v8f d = __builtin_amdgcn_wmma_f32_16x16x4_f32(
      /*neg_a=*/false, a, /*neg_b=*/false, b,
      /*c_mod=*/(short)0, c, /*reuse_a=*/false, /*reuse_b=*/false);
  r[0] = __shfl(d[0], 0, 32); r[1] = __shfl(d[0], 1, 32); r[2] = __shfl(d[0], 2, 32);
  r[3] = __shfl(d[1], 0, 32); r[4] = __shfl(d[1], 1, 32); r[5] = __shfl(d[1], 2, 32);
  r[6] = __shfl(d[2], 0, 32); r[7] = __shfl(d[2], 1, 32); r[8] = __shfl(d[2], 2, 32);
}

// ---------------------------------------------------------------------------
// Kernel 1: per-batch weights -> generator -> expm (Taylor 10 + 6 squarings)
// -> pixel-space affine coefficients [ax,bx,cx,ay,by,cy] in d_ws.
// One wave (32 threads) per batch element. 256 waves total: negligible cost.
// ---------------------------------------------------------------------------
__global__ __launch_bounds__(32)
void expm_theta_kernel(const float* __restrict__ width,
                       const float* __restrict__ rand_u,
                       float* __restrict__ coef) {
  const int b = blockIdx.x;

  // wave-uniform scaled generator G / 2^6
  float g[9];
  {
    float wgt[6];
#pragma unroll
    for (int i = 0; i < 6; ++i) {
      float wi = width[i];
      float sp = (wi > 20.0f) ? wi : log1pf(expf(wi));      // softplus
      wgt[i] = rand_u[b * 6 + i] * sp - sp * 0.5f;
    }
    const float TS = 0.1f;     // TRANS_SCALE
    g[0] = wgt[3] + wgt[4];    // (0,0)
    g[1] = -wgt[2] + wgt[5];   // (0,1)
    g[2] = wgt[0] * TS;        // (0,2)
    g[3] = wgt[2] + wgt[5];    // (1,0)
    g[4] = wgt[3] - wgt[4];    // (1,1)
    g[5] = wgt[1] * TS;        // (1,2)
    g[6] = 0.0f; g[7] = 0.0f; g[8] = 0.0f;
    const float s = 1.0f / 64.0f;   // 2^-squarings
#pragma unroll
    for (int k = 0; k < 9; ++k) g[k] *= s;
  }

  float out[9]  = {1.f, 0.f, 0.f, 0.f, 1.f, 0.f, 0.f, 0.f, 1.f};
  float term[9] = {1.f, 0.f, 0.f, 0.f, 1.f, 0.f, 0.f, 0.f, 1.f};

#pragma unroll
  for (int i = 1; i <= 10; ++i) {              // Taylor: term = term @ G / i
    float nt[9];
    wave_mm3(term, g, nt);
    const float inv = 1.0f / (float)i;
#pragma unroll
    for (int k = 0; k < 9; ++k) { term[k] = nt[k] * inv; out[k] += term[k]; }
  }
#pragma unroll
  for (int s2 = 0; s2 < 6; ++s2) {             // squarings: out = out @ out
    float nt[9];
    wave_mm3(out, out, nt);
#pragma unroll
    for (int k = 0; k < 9; ++k) out[k] = nt[k];
  }

  if (threadIdx.x == 0) {
    const float t00 = out[0], t01 = out[1], t02 = out[2];
    const float t10 = out[3], t11 = out[4], t12 = out[5];
    const float Wm1 = (float)(IMG_W - 1), Hm1 = (float)(IMG_H - 1);
    // ix = ax*px + bx*py + cx  (pixel units, align_corners=True folded in)
    float* p = coef + b * 6;
    p[0] = t00;
    p[1] = t01 * (Wm1 / Hm1);
    p[2] = 0.5f * Wm1 * (t02 + 1.0f - t00 - t01);
    p[3] = t10 * (Hm1 / Wm1);
    p[4] = t11;
    p[5] = 0.5f * Hm1 * (t12 + 1.0f - t10 - t11);
  }
}

// ---------------------------------------------------------------------------
// Kernel 2: bilinear grid sample, zeros padding. One thread = 4 x-pixels x 3 ch.
// Memory-bound: ~308 MB HBM traffic (~13 us at 23.3 TB/s). The 154 MB input
// fits in the 192 MB L2, so bilinear re-reads are L2 hits; output stores are
// non-temporal so they don't evict the gather working set from L2.
// ---------------------------------------------------------------------------
__global__ __launch_bounds__(BLK)
void sample_kernel(const float* __restrict__ img,
                   const float* __restrict__ coef,
                   float* __restrict__ outp) {
  const int b = blockIdx.y;                 // uniform -> scalar loads for coef
  const float* cf = coef + b * 6;
  const float ax = cf[0], bx = cf[1], cx = cf[2];
  const float ay = cf[3], by = cf[4], cy = cf[5];

  const int id = blockIdx.x * BLK + threadIdx.x;   // 0 .. 12543
  const int y  = id / XQ;
  const int x0 = (id - y * XQ) * PXT;

  const float py  = (float)y;
  const float ix0f = ax * (float)x0 + bx * py + cx;
  const float iy0f = ay * (float)x0 + by * py + cy;

  const float* src = img  + (size_t)b * IMG_C * HW;
  float*       dst = outp + (size_t)b * IMG_C * HW + (size_t)y * IMG_W + x0;

  float res[IMG_C][PXT];
#pragma unroll
  for (int p = 0; p < PXT; ++p) {
    const float ix = ix0f + ax * (float)p;
    const float iy = iy0f + ay * (float)p;
    const float fx = floorf(ix), fy = floorf(iy);
    const float wx1 = ix - fx, wy1 = iy - fy;
    const float wx0 = 1.0f - wx1, wy0 = 1.0f - wy1;
    const int jx0 = (int)fx, jy0 = (int)fy;
    const int jx1 = jx0 + 1, jy1 = jy0 + 1;
    const bool vx0 = (unsigned)jx0 < (unsigned)IMG_W;
    const bool vx1 = (unsigned)jx1 < (unsigned)IMG_W;
    const bool vy0 = (unsigned)jy0 < (unsigned)IMG_H;
    const bool vy1 = (unsigned)jy1 < (unsigned)IMG_H;
    const int cx0 = min(max(jx0, 0), IMG_W - 1);
    const int cx1 = min(max(jx1, 0), IMG_W - 1);
    const int cy0 = min(max(jy0, 0), IMG_H - 1);
    const int cy1 = min(max(jy1, 0), IMG_H - 1);
    const float w00 = wy0 * wx0 * (float)(vy0 && vx0);
    const float w01 = wy0 * wx1 * (float)(vy0 && vx1);
    const float w10 = wy1 * wx0 * (float)(vy1 && vx0);
    const float w11 = wy1 * wx1 * (float)(vy1 && vx1);
    const int o00 = cy0 * IMG_W + cx0, o01 = cy0 * IMG_W + cx1;
    const int o10 = cy1 * IMG_W + cx0, o11 = cy1 * IMG_W + cx1;
#pragma unroll
    for (int c = 0; c < IMG_C; ++c) {
      const float* s = src + c * HW;
      res[c][p] = s[o00] * w00 + s[o01] * w01 + s[o10] * w10 + s[o11] * w11;
    }
  }
#pragma unroll
  for (int c = 0; c < IMG_C; ++c) {
    v4f v = { res[c][0], res[c][1], res[c][2], res[c][3] };
    // Non-temporal b128 store: output is never re-read; keep L2 for the input.
    __builtin_nontemporal_store(v, (v4f*)(dst + c * HW));
  }
}

extern "C" void kernel_launch(void* const* d_in, const int* in_sizes, int n_in,
                              void* d_out, int out_size, void* d_ws, size_t ws_size,
                              hipStream_t stream) {
  const float* x      = (const float*)d_in[0];   // (B,3,224,224) f32
  const float* width  = (const float*)d_in[1];   // (6,) f32
  const float* rand_u = (const float*)d_in[2];   // (B,6) f32
  float* out  = (float*)d_out;
  float* coef = (float*)d_ws;                    // B*6 floats scratch

  const int B = in_sizes[2] / 6;

  expm_theta_kernel<<<B, 32, 0, stream>>>(width, rand_u, coef);
  dim3 grid(NBLK, B);
  sample_kernel<<<grid, BLK, 0, stream>>>(x, coef, out);
}